// IntegerNeuron_84842783965742
// MI455X (gfx1250) — compile-verified
//
#include <hip/hip_runtime.h>

// Problem geometry (fixed by the reference).
#define T_STEPS 8
#define B_N 32
#define C_CH 128
#define H_N 32
#define W_N 32
#define PLANE ((size_t)B_N * C_CH * H_N * W_N)  // 4,194,304 floats per time plane
#define EPS_F 1e-12f
#define BLOCK 256

typedef float v4f __attribute__((ext_vector_type(4)));

// One thread owns one float4 (4 consecutive W elements -> single channel).
// Per block: stage T_STEPS planes x (BLOCK float4) = 8 * 4KB = 32KB LDS via
// gfx1250 async global->LDS DMA (non-temporal), then a software-pipelined
// sequential spiking scan: plane t is consumed under s_wait_asynccnt <= 7-t
// (async loads of one wave complete in order), with NT b128 spike stores.
__global__ __launch_bounds__(BLOCK) void integer_neuron_scan_kernel(
    const float* __restrict__ x,          // [T,B,C,H,W]
    const float* __restrict__ prev_scale, // [C]
    const float* __restrict__ prev_bias,  // [C]
    const float* __restrict__ vth_p,      // scalar
    const int* __restrict__ tau_p,        // scalar (python int)
    const int* __restrict__ first_p,      // scalar (python int)
    float* __restrict__ out)              // [T,B,C,H,W]
{
    __shared__ v4f stage[T_STEPS * BLOCK];  // 32 KB

    const unsigned g = blockIdx.x * BLOCK + threadIdx.x;  // float4 index in a plane
    const size_t e = (size_t)g * 4u;                      // element index in a plane
    const int c = (g >> 8) & (C_CH - 1);                  // (e / 1024) % 128

    // ---- Issue all T async global->LDS b128 transfers (ASYNCcnt path) ----
    // Single-touch input: non-temporal load hint keeps L2 clean.
    // LDS byte offset = low 32 bits of the generic pointer to shared memory.
#pragma unroll
    for (int t = 0; t < T_STEPS; ++t) {
        unsigned lds_off =
            (unsigned)(uintptr_t)(&stage[t * BLOCK + threadIdx.x]);
        const float* gp = x + (size_t)t * PLANE + e;
        asm volatile("global_load_async_to_lds_b128 %0, %1, off th:TH_LOAD_NT"
                     :
                     : "v"(lds_off), "v"(gp)
                     : "memory");
    }

    // ---- Per-channel constants (overlap with the in-flight async DMA) ----
    const float tau_f = (float)(*tau_p);
    const float scale = prev_scale[c];
    const float inv = tau_f / (scale + EPS_F);
    const float bias_s = rintf(prev_bias[c] * inv);   // JAX round = half-to-even
    const float vth_s = rintf((*vth_p) * inv);
    const float mult = (*first_p) ? 1.0f : tau_f;     // drive = x or x*tau

    // ---- Pipelined soft-reset spiking scan over T ----
    // Async loads complete in order: plane t is resident once ASYNCcnt <= 7-t.
    v4f mem = (v4f)0.0f;

#define SCAN_STEP(T_IDX)                                                       \
    do {                                                                       \
        asm volatile("s_wait_asynccnt %0"                                      \
                     :                                                         \
                     : "n"(T_STEPS - 1 - (T_IDX))                              \
                     : "memory");                                              \
        v4f xv = stage[(T_IDX) * BLOCK + threadIdx.x]; /* ds_load_b128 */      \
        v4f s;                                                                 \
        mem.x += xv.x * mult + bias_s;                                         \
        s.x = (mem.x >= vth_s) ? 1.0f : 0.0f;                                  \
        mem.x -= s.x * vth_s;                                                  \
        mem.y += xv.y * mult + bias_s;                                         \
        s.y = (mem.y >= vth_s) ? 1.0f : 0.0f;                                  \
        mem.y -= s.y * vth_s;                                                  \
        mem.z += xv.z * mult + bias_s;                                         \
        s.z = (mem.z >= vth_s) ? 1.0f : 0.0f;                                  \
        mem.z -= s.z * vth_s;                                                  \
        mem.w += xv.w * mult + bias_s;                                         \
        s.w = (mem.w >= vth_s) ? 1.0f : 0.0f;                                  \
        mem.w -= s.w * vth_s;                                                  \
        __builtin_nontemporal_store(                                           \
            s, reinterpret_cast<v4f*>(out + (size_t)(T_IDX) * PLANE + e));     \
    } while (0)

    SCAN_STEP(0);
    SCAN_STEP(1);
    SCAN_STEP(2);
    SCAN_STEP(3);
    SCAN_STEP(4);
    SCAN_STEP(5);
    SCAN_STEP(6);
    SCAN_STEP(7);
#undef SCAN_STEP
}

extern "C" void kernel_launch(void* const* d_in, const int* in_sizes, int n_in,
                              void* d_out, int out_size, void* d_ws, size_t ws_size,
                              hipStream_t stream) {
    (void)in_sizes; (void)n_in; (void)d_ws; (void)ws_size; (void)out_size;

    const float* x          = (const float*)d_in[0];
    const float* prev_scale = (const float*)d_in[1];
    const float* prev_bias  = (const float*)d_in[2];
    const float* vth        = (const float*)d_in[3];
    const int*   tau        = (const int*)d_in[4];
    const int*   first      = (const int*)d_in[5];
    float*       out        = (float*)d_out;

    const unsigned n_vec4 = (unsigned)(PLANE / 4);  // 1,048,576 threads
    dim3 grid(n_vec4 / BLOCK);                      // 4096 blocks
    dim3 block(BLOCK);
    integer_neuron_scan_kernel<<<grid, block, 0, stream>>>(
        x, prev_scale, prev_bias, vth, tau, first, out);
}